// Linear_6983616824492
// MI455X (gfx1250) — compile-verified
//
#include <hip/hip_runtime.h>

// CDNA5 / gfx1250 wave32 WMMA types
typedef float v2f __attribute__((ext_vector_type(2)));
typedef float v8f __attribute__((ext_vector_type(8)));

#define PITCH 1668   // 1664 padded to multiple of 4; pitch % 64 == 4 -> conflict-free LDS patterns
#define NROW  16
#define ROWF  1664   // floats per batch row

// One n-tile task: computes ALL `DIM` interleaved 16x16 output tiles that share the
// same weight tile W[:, nt*16 .. nt*16+16).  B is loaded once per K-step and reused
// across the DIM independent accumulators -> DIM x fewer VMEM loads, and the DIM
// WMMAs per K-step are independent (no D->C chain), improving matrix-pipe ILP.
//
// A-layout (32-bit 16x4): lanes 0-15: M=lane, VGPR0=K+0, VGPR1=K+1; lanes 16-31: K+2/K+3.
// B-layout (32-bit 4x16): lanes 0-15: N=lane, rows K+0/K+1; lanes 16-31: rows K+2/K+3.
// D-layout: VGPR v -> row M = v + 8*(lane>=16), col N = lane&15.
template<int K, int DIM, int MULN>
__device__ __forceinline__ void tile_task(const float* __restrict__ lds_in,
                                          float* __restrict__ lds_out,
                                          const float* __restrict__ W,
                                          float pw, int nt,
                                          int inoff, int outbase, int lane)
{
    const int m    = lane & 15;
    const int half = lane >> 4;     // 0: lanes 0-15, 1: lanes 16-31
    const int kofs = half << 1;     // +0 or +2 in K per ISA layout

    const float* __restrict__ arow = lds_in + m * PITCH + inoff + kofs * DIM;
    const float* __restrict__ wp   = W + nt * 16 + m + kofs * MULN;

    const v8f vzero = {0.f, 0.f, 0.f, 0.f, 0.f, 0.f, 0.f, 0.f};

    if constexpr (DIM == 1) {
        // Two split accumulators to break the D->C serialization; sum at the end.
        v8f acc0 = vzero, acc1 = vzero;
#pragma unroll 2
        for (int k = 0; k < K; k += 8) {
            v2f a0, b0, a1, b1;
            b0.x = wp[k * MULN];           b0.y = wp[(k + 1) * MULN];
            b1.x = wp[(k + 4) * MULN];     b1.y = wp[(k + 5) * MULN];
            a0.x = arow[k];                a0.y = arow[k + 1];
            a1.x = arow[k + 4];            a1.y = arow[k + 5];
            acc0 = __builtin_amdgcn_wmma_f32_16x16x4_f32(
                       false, a0, false, b0, (short)0, acc0, false, false);
            acc1 = __builtin_amdgcn_wmma_f32_16x16x4_f32(
                       false, a1, false, b1, (short)0, acc1, false, false);
        }
        v8f acc = acc0 + acc1;
        float* __restrict__ orow = lds_out + (half * 8) * PITCH
                                 + outbase + (nt * 16 + m);
#pragma unroll
        for (int v = 0; v < 8; ++v)
            orow[v * PITCH] = acc[v] * pw;
    } else {
        v8f acc[DIM];
#pragma unroll
        for (int d = 0; d < DIM; ++d) acc[d] = vzero;

#pragma unroll 2
        for (int k = 0; k < K; k += 4) {
            v2f b;
            b.x = wp[k * MULN];            // W[k+kofs,   nt*16+m]  (L2-resident)
            b.y = wp[k * MULN + MULN];     // W[k+kofs+1, nt*16+m]
#pragma unroll
            for (int d = 0; d < DIM; ++d) {
                v2f a;
                a.x = arow[k * DIM + d];           // x[b0+m, inoff + (k+kofs)*DIM + d]
                a.y = arow[k * DIM + DIM + d];     //           ... (k+kofs+1)
                acc[d] = __builtin_amdgcn_wmma_f32_16x16x4_f32(
                             false, a, false, b, (short)0, acc[d], false, false);
            }
        }

        float* __restrict__ obase = lds_out + (half * 8) * PITCH
                                  + outbase + (nt * 16 + m) * DIM;
#pragma unroll
        for (int d = 0; d < DIM; ++d)
#pragma unroll
            for (int v = 0; v < 8; ++v)
                obase[v * PITCH + d] = acc[d][v] * pw;
    }
}

__global__ void __launch_bounds__(256, 1)
e3nn_linear_wmma_kernel(const float* __restrict__ x,
                        const float* __restrict__ w0,
                        const float* __restrict__ w1,
                        const float* __restrict__ w2,
                        float* __restrict__ out)
{
    extern __shared__ float smem[];
    float* lds_in  = smem;                 // 16 x PITCH
    float* lds_out = smem + NROW * PITCH;  // 16 x PITCH

    const int    tid  = threadIdx.x;
    const int    lane = tid & 31;
    const int    wave = tid >> 5;
    const size_t b0   = (size_t)blockIdx.x * NROW;   // 50000 = 3125 * 16, exact

    // ---- Stage input slab: 16 rows x 1664 f32, coalesced float4 (b128) loads ----
    {
        const int r  = tid >> 4;     // 0..15 row
        const int c0 = tid & 15;     // 16 threads per row
        const float4* __restrict__ srow = (const float4*)(x + (b0 + r) * ROWF);
        float4* __restrict__       drow = (float4*)(lds_in + r * PITCH); // 16B aligned
#pragma unroll
        for (int j = 0; j < 26; ++j)                   // 26*16 = 416 float4 = 1664 floats
            drow[c0 + j * 16] = srow[c0 + j * 16];
    }
    __syncthreads();

    // ---- 40 wave-tasks: 16 (block0) + 16 (block1, all 3 i's) + 8 (block2, all 5 i's)
    // Perfectly balanced: every wave gets 2+2+1 tasks = 672 WMMAs.
    for (int t = wave; t < 40; t += 8) {
        if (t < 16) {
            // block0: mul=256, dim=1, pw=1/sqrt(256)
            tile_task<256, 1, 256>(lds_in, lds_out, w0, 0.0625f, t, 0, 0, lane);
        } else if (t < 32) {
            // block1: mul=256, dim=3, x cols [256,1024)
            tile_task<256, 3, 256>(lds_in, lds_out, w1, 0.0625f, t - 16, 256, 256, lane);
        } else {
            // block2: mul=128, dim=5, x cols [1024,1664), pw=1/sqrt(128)
            tile_task<128, 5, 128>(lds_in, lds_out, w2, 0.08838834764831845f, t - 32, 1024, 1024, lane);
        }
    }
    __syncthreads();

    // ---- Write back output slab: coalesced float4 (b128) stores ----
    {
        const int r  = tid >> 4;
        const int c0 = tid & 15;
        const float4* __restrict__ srow = (const float4*)(lds_out + r * PITCH);
        float4* __restrict__       drow = (float4*)(out + (b0 + r) * ROWF);
#pragma unroll
        for (int j = 0; j < 26; ++j)
            drow[c0 + j * 16] = srow[c0 + j * 16];
    }
}

extern "C" void kernel_launch(void* const* d_in, const int* in_sizes, int n_in,
                              void* d_out, int out_size, void* d_ws, size_t ws_size,
                              hipStream_t stream)
{
    (void)in_sizes; (void)n_in; (void)out_size; (void)d_ws; (void)ws_size;
    const float* x  = (const float*)d_in[0];
    const float* w0 = (const float*)d_in[1];
    const float* w1 = (const float*)d_in[2];
    const float* w2 = (const float*)d_in[3];
    float* out = (float*)d_out;

    const int    nblocks = 50000 / NROW;                             // 3125
    const size_t shmem   = (size_t)2 * NROW * PITCH * sizeof(float); // 213504 B < 320 KB/WGP

    e3nn_linear_wmma_kernel<<<nblocks, 256, shmem, stream>>>(x, w0, w1, w2, out);
}